// MyModel_61933428409029
// MI455X (gfx1250) — compile-verified
//
#include <hip/hip_runtime.h>
#include <stdint.h>

// Streaming, memory-bound kernel: 304 MB total traffic -> ~13 us floor at
// 23.3 TB/s. Uses CDNA5 async global->LDS loads (ASYNCcnt) for the input
// tile, conflict-free stride-11 LDS reads, and 32B-aligned b128 stores.

#define ROWS_PER_BLOCK 256
#define FPR 11  // floats per input row

__global__ __launch_bounds__(ROWS_PER_BLOCK)
void contact_kkt_kernel(const float* __restrict__ x,
                        const float* __restrict__ A1,
                        const float* __restrict__ b1,
                        const float* __restrict__ A2,
                        const float* __restrict__ b2,
                        float* __restrict__ out,
                        int B)
{
    __shared__ __align__(16) float tile[ROWS_PER_BLOCK * FPR]; // 11264 B

    const int tid = threadIdx.x;
    const int block_row0 = (int)blockIdx.x * ROWS_PER_BLOCK;

    int rows = B - block_row0;
    if (rows > ROWS_PER_BLOCK) rows = ROWS_PER_BLOCK;
    const int total_floats = rows * FPR;
    const int n128 = total_floats >> 2;           // full 16B chunks

    // Low 32 bits of a generic pointer into LDS == LDS byte offset.
    const unsigned lds_base = (unsigned)(uintptr_t)&tile[0];
    const uint64_t src_base = (uint64_t)(const void*)(x + (size_t)block_row0 * FPR);

    // --- Async copy global -> LDS, coalesced 16B chunks (CDNA5 path) ---
    for (int t = tid; t < n128; t += ROWS_PER_BLOCK) {
        unsigned lds_off = lds_base + (unsigned)(t << 4);
        uint64_t gaddr   = src_base + ((uint64_t)t << 4);
        asm volatile("global_load_async_to_lds_b128 %0, %1, off"
                     :: "v"(lds_off), "v"(gaddr) : "memory");
    }
    // Ragged tail (only possible in a partial last block): b32 async loads.
    for (int t = (n128 << 2) + tid; t < total_floats; t += ROWS_PER_BLOCK) {
        unsigned lds_off = lds_base + (unsigned)(t << 2);
        uint64_t gaddr   = src_base + ((uint64_t)t << 2);
        asm volatile("global_load_async_to_lds_b32 %0, %1, off"
                     :: "v"(lds_off), "v"(gaddr) : "memory");
    }
    asm volatile("s_wait_asynccnt 0x0" ::: "memory");
    __syncthreads();

    const int row = block_row0 + tid;
    if (row < B) {
        // Stride-11 (odd) LDS reads: bank-conflict-free across a wave32.
        const float* r = &tile[tid * FPR];
        const float p1x = r[0], p1y = r[1];
        const float p2x = r[2], p2y = r[3];
        const float l1x = r[4], l1y = r[5];
        const float l2x = r[6], l2y = r[7];
        const float qx  = r[8], qy  = r[9];
        const float th  = r[10];

        float sn, cs;
        sincosf(th, &sn, &cs);

        // Uniform 2x2 constants (L2-cached uniform loads).
        const float a100 = A1[0], a101 = A1[1], a110 = A1[2], a111 = A1[3];
        const float b1x  = b1[0], b1y  = b1[1];
        const float a200 = A2[0], a201 = A2[1], a210 = A2[2], a211 = A2[3];
        const float b2x  = b2[0], b2y  = b2[1];

        // box1_A = A1 @ rot_inv, rot_inv = [[c, s], [-s, c]]
        const float BA00 = a100 * cs - a101 * sn;
        const float BA01 = a100 * sn + a101 * cs;
        const float BA10 = a110 * cs - a111 * sn;
        const float BA11 = a110 * sn + a111 * cs;

        const float bb0 = b1x + BA00 * qx + BA01 * qy;
        const float bb1 = b1y + BA10 * qx + BA11 * qy;

        const float dpx = p1x - p2x;
        const float dpy = p1y - p2y;

        // k_p1 = 2*dp + box1_A^T @ lam1
        const float kp1x =  2.0f * dpx + BA00 * l1x + BA10 * l1y;
        const float kp1y =  2.0f * dpy + BA01 * l1x + BA11 * l1y;
        // k_p2 = -2*dp + A2^T @ lam2
        const float kp2x = -2.0f * dpx + a200 * l2x + a210 * l2y;
        const float kp2y = -2.0f * dpy + a201 * l2x + a211 * l2y;
        // k_l1 = lam1 * (box1_A @ p1 - box1_b)
        const float kl1x = l1x * (BA00 * p1x + BA01 * p1y - bb0);
        const float kl1y = l1y * (BA10 * p1x + BA11 * p1y - bb1);
        // k_l2 = lam2 * (A2 @ p2 - b2)
        const float kl2x = l2x * (a200 * p2x + a201 * p2y - b2x);
        const float kl2y = l2y * (a210 * p2x + a211 * p2y - b2y);

        // 32 B per row, 16B-aligned -> two b128 stores, fully coalesced.
        float4* o = (float4*)(out + (size_t)row * 8);
        o[0] = make_float4(kp1x, kp1y, kp2x, kp2y);
        o[1] = make_float4(kl1x, kl1y, kl2x, kl2y);
    }
}

extern "C" void kernel_launch(void* const* d_in, const int* in_sizes, int n_in,
                              void* d_out, int out_size, void* d_ws, size_t ws_size,
                              hipStream_t stream) {
    const float* x  = (const float*)d_in[0];
    const float* A1 = (const float*)d_in[1];
    const float* b1 = (const float*)d_in[2];
    const float* A2 = (const float*)d_in[3];
    const float* b2 = (const float*)d_in[4];
    float* out = (float*)d_out;

    const int B = in_sizes[0] / FPR;
    const int grid = (B + ROWS_PER_BLOCK - 1) / ROWS_PER_BLOCK;

    contact_kkt_kernel<<<dim3(grid), dim3(ROWS_PER_BLOCK), 0, stream>>>(
        x, A1, b1, A2, b2, out, B);
}